// BlockConvolution_61220463837575
// MI455X (gfx1250) — compile-verified
//
#include <hip/hip_runtime.h>

// ---------------------------------------------------------------------------
// BlockConvolution on MI455X (gfx1250): recast as fp32 WMMA GEMM
//   A[8192 x 3584] = seq blocks (rows k=0..6 of each 8-row block, contiguous)
//   B[3584 x 3584] = masked W repacked (k*512+e major, i*512+o minor)
//   C = A*B + bias -> scattered into rows 1..7 of each output block
// Memory-bound (~440MB @ 23.3TB/s); fp32 WMMA keeps reference precision.
// ---------------------------------------------------------------------------

typedef float v2f __attribute__((ext_vector_type(2)));
typedef float v8f __attribute__((ext_vector_type(8)));

#define EDIM 512
#define NDIM 3584            // 7*512 : both GEMM N (io) and K (ke) extents
#define MDIM 8192            // 16*512 block instances
#define SROW 4096            // floats per block row in seq/out (8*512)

#define BM 128
#define BN 128
#define BKT 32
#define AS_STRIDE 36         // 32 + 4 pad: keeps float2 frag reads conflict-free
#define BS_STRIDE 136        // 128 + 8 pad: half-wave rows land on disjoint banks

// ---- Kernel 1: repack + mask W[i,o,e,k] -> Bp[(k*512+e)][ (i*512+o) ] -------
// W flat index  = io*3584 + (e*7 + k)   (io = i*512+o)
// Bp flat index = (k*512+e)*3584 + io
// Tiled 32x32 LDS transpose: coalesced 128B reads AND writes.
__global__ __launch_bounds__(256) void bc_repack_w(const float* __restrict__ W,
                                                   float* __restrict__ Bp) {
    __shared__ float tile[32][33];
    const int t   = threadIdx.x;
    const int io0 = (blockIdx.x % (NDIM / 32)) * 32;
    const int f0  = (blockIdx.x / (NDIM / 32)) * 32;
#pragma unroll
    for (int j = 0; j < 4; ++j) {
        int r = j * 8 + (t >> 5);          // io row within tile
        int c = t & 31;                    // f col within tile
        tile[r][c] = W[(size_t)(io0 + r) * NDIM + (f0 + c)];
    }
    __syncthreads();
#pragma unroll
    for (int j = 0; j < 4; ++j) {
        int fi = j * 8 + (t >> 5);
        int ii = t & 31;
        int f  = f0 + fi;
        int k  = f % 7;
        int e  = f / 7;
        int io = io0 + ii;
        int i  = io >> 9;                  // conv index
        float v = (k <= i) ? tile[ii][fi] : 0.0f;
        Bp[(size_t)(k * EDIM + e) * NDIM + io] = v;
    }
}

// ---- Kernel 2: zero row 0 of every output block (d_out is poisoned) --------
__global__ __launch_bounds__(256) void bc_zero_slot0(float* __restrict__ out) {
    size_t t  = (size_t)blockIdx.x * 256 + threadIdx.x;   // over 8192*128 float4
    size_t nb = t >> 7;
    size_t o4 = (t & 127) << 2;
    float4 z; z.x = 0.f; z.y = 0.f; z.z = 0.f; z.w = 0.f;
    *(float4*)(out + nb * SROW + o4) = z;
}

// ---- Kernel 3: tiled fp32 WMMA GEMM + bias + block scatter -----------------
__global__ __launch_bounds__(256) void bc_gemm(const float* __restrict__ Aseq,
                                               const float* __restrict__ Bp,
                                               const float* __restrict__ bias,
                                               float* __restrict__ out) {
    __shared__ float As[BM * AS_STRIDE];   // [m][k], padded
    __shared__ float Bs[BKT * BS_STRIDE];  // [k][n], padded

    const int t    = threadIdx.x;
    const int lane = t & 31;
    const int w    = t >> 5;
    const int wm   = w & 3;                // wave M slot: rows wm*32..+31
    const int wn   = w >> 2;               // wave N slot: cols wn*64..+63
    const int m0   = blockIdx.y * BM;
    const int n0   = blockIdx.x * BN;

    const int ln16  = lane & 15;
    const int khalf = (lane >> 4) << 1;    // 0 (lanes 0-15) or 2 (lanes 16-31)

    v8f acc[2][4];
#pragma unroll
    for (int mt = 0; mt < 2; ++mt)
#pragma unroll
        for (int nt = 0; nt < 4; ++nt)
            acc[mt][nt] = (v8f)0.0f;

    for (int kt = 0; kt < NDIM; kt += BKT) {
        // --- A tile 128x32: coalesced float4 global loads, padded LDS store
        {
            int c4 = (t & 7) << 2;
            int rb = t >> 3;               // 0..31
#pragma unroll
            for (int j = 0; j < 4; ++j) {
                int r = rb + j * 32;
                float4 v = *(const float4*)(Aseq + (size_t)(m0 + r) * SROW + kt + c4);
                *(float4*)(As + r * AS_STRIDE + c4) = v;
            }
        }
        // --- B tile 32x128: coalesced float4 global loads, padded LDS store
        {
            int c4 = (t & 31) << 2;
            int kb = t >> 5;               // 0..7
#pragma unroll
            for (int j = 0; j < 4; ++j) {
                int kr = kb + j * 8;
                float4 v = *(const float4*)(Bp + (size_t)(kt + kr) * NDIM + n0 + c4);
                *(float4*)(Bs + kr * BS_STRIDE + c4) = v;
            }
        }
        __syncthreads();

#pragma unroll
        for (int kk = 0; kk < BKT / 4; ++kk) {
            const int koff = kk * 4 + khalf;
            v2f af[2], bf[4];
#pragma unroll
            for (int mt = 0; mt < 2; ++mt) {
                int m = wm * 32 + mt * 16 + ln16;
                af[mt] = *(const v2f*)(As + m * AS_STRIDE + koff);
            }
#pragma unroll
            for (int nt = 0; nt < 4; ++nt) {
                int n = wn * 64 + nt * 16 + ln16;
                bf[nt].x = Bs[koff * BS_STRIDE + n];
                bf[nt].y = Bs[(koff + 1) * BS_STRIDE + n];
            }
#pragma unroll
            for (int mt = 0; mt < 2; ++mt)
#pragma unroll
                for (int nt = 0; nt < 4; ++nt)
                    acc[mt][nt] = __builtin_amdgcn_wmma_f32_16x16x4_f32(
                        false, af[mt], false, bf[nt],
                        (short)0, acc[mt][nt], false, false);
        }
        __syncthreads();
    }

    // --- Epilogue: bias add + scatter to rows 1..7 of each 8-row block ------
#pragma unroll
    for (int nt = 0; nt < 4; ++nt) {
        int n_g = n0 + wn * 64 + nt * 16 + ln16;       // io = i*512 + o
        float bv = bias[n_g];
#pragma unroll
        for (int mt = 0; mt < 2; ++mt) {
#pragma unroll
            for (int r = 0; r < 8; ++r) {
                // C layout: VGPR r -> M row r (lanes 0-15) / r+8 (lanes 16-31)
                int m_g = m0 + wm * 32 + mt * 16 + r + ((lane >> 4) << 3);
                out[(size_t)m_g * SROW + EDIM + n_g] = acc[mt][nt][r] + bv;
            }
        }
    }
}

extern "C" void kernel_launch(void* const* d_in, const int* in_sizes, int n_in,
                              void* d_out, int out_size, void* d_ws, size_t ws_size,
                              hipStream_t stream) {
    const float* seq  = (const float*)d_in[0];   // [16,4096,512] f32
    const float* W    = (const float*)d_in[1];   // [7,512,512,7] f32
    const float* bias = (const float*)d_in[2];   // [7,512] f32
    float* out = (float*)d_out;
    float* Bp  = (float*)d_ws;                   // needs 3584*3584*4 = 51.4 MB

    // 1) repack/mask W -> Bp   (112 x 112 tiles of 32x32)
    bc_repack_w<<<(NDIM / 32) * (NDIM / 32), 256, 0, stream>>>(W, Bp);
    // 2) zero slot-0 rows of output
    bc_zero_slot0<<<(MDIM * (EDIM / 4)) / 256, 256, 0, stream>>>(out);
    // 3) WMMA GEMM + bias + scatter
    dim3 grid(NDIM / BN, MDIM / BM);             // 28 x 64
    bc_gemm<<<grid, 256, 0, stream>>>(seq, Bp, bias, out);
}